// Resize3D_52304111731438
// MI455X (gfx1250) — compile-verified
//
#include <hip/hip_runtime.h>

// Nearest-neighbor 3D resize (2,64,128,128,16) f32 -> (2,96,192,192,16) f32.
// All axes use scale 2/3 with half-pixel centers: src = (2*dst + 1) / 3.
// Pure bandwidth kernel: ~0.59 GB traffic -> ~25us at 23.3 TB/s on MI455X.
// CDNA5 paths used: GLOBAL_LOAD_ASYNC_TO_LDS_B128 (ASYNCcnt), s_wait_asynccnt,
// ds_load_b128, non-temporal global_store_b128.

typedef float f4  __attribute__((ext_vector_type(4)));
typedef int   v4i __attribute__((ext_vector_type(4)));

#define AS1 __attribute__((address_space(1)))
#define AS3 __attribute__((address_space(3)))

typedef AS1 v4i* g_v4i_ptr;  // global-space int4*   (builtin arg 1)
typedef AS3 v4i* l_v4i_ptr;  // LDS-space int4*      (builtin arg 2)

#if defined(__has_builtin)
#if __has_builtin(__builtin_amdgcn_global_load_async_to_lds_b128)
#define USE_ASYNC_BUILTIN 1
#endif
#if __has_builtin(__builtin_amdgcn_s_wait_asynccnt)
#define USE_WAIT_BUILTIN 1
#endif
#endif

__device__ __forceinline__ void async_f4_to_lds(const f4* gptr, f4* lds_generic) {
#if defined(USE_ASYNC_BUILTIN)
    // Erase const/addrspace/pointee via integer casts, then rebuild the exact
    // parameter types. AS1 pointers are 64-bit; AS3 pointers are 32-bit and
    // the low 32 bits of a flat shared address are the LDS byte offset.
    g_v4i_ptr g = (g_v4i_ptr)(uintptr_t)gptr;
    l_v4i_ptr l = (l_v4i_ptr)(unsigned)(uintptr_t)lds_generic;
    __builtin_amdgcn_global_load_async_to_lds_b128(g, l, /*offset=*/0, /*cpol=*/0);
#else
    // Fallback: raw CDNA5 async-copy instruction (VDST = wave-relative LDS
    // byte address, VADDR = 64-bit global address).  CDNA5 ISA.
    unsigned lds_off = (unsigned)(uintptr_t)lds_generic;
    unsigned long long gaddr = (unsigned long long)(uintptr_t)gptr;
    asm volatile("global_load_async_to_lds_b128 %0, %1, off"
                 :
                 : "v"(lds_off), "v"(gaddr)
                 : "memory");
#endif
}

__device__ __forceinline__ void wait_async_zero() {
#if defined(USE_WAIT_BUILTIN)
    __builtin_amdgcn_s_wait_asynccnt(0);
#else
    asm volatile("s_wait_asynccnt 0" ::: "memory");
#endif
}

// Grid: one block per output (b, d_out, h_out) row; 2*96*192 = 36864 blocks.
// Block: 256 threads (8 wave32).
//   Stage: async-copy the needed input row (128 voxels * 16ch * 4B = 8KB,
//          i.e. 512 float4) into LDS -> 2 async b128 loads per thread.
//   Emit:  output row is 192 voxels * 16ch = 768 float4 -> 3 coalesced
//          non-temporal b128 stores per thread, gathering w_in from LDS.
__global__ __launch_bounds__(256) void Resize3D_52304111731438_kernel(
    const float* __restrict__ in, float* __restrict__ out) {
    __shared__ f4 tile[512];  // one input row: 128 voxels * 4 f4

    const unsigned t   = threadIdx.x;
    const unsigned bid = blockIdx.x;           // = (b*96 + d_out)*192 + h_out
    const unsigned ho  = bid % 192u;
    const unsigned tmp = bid / 192u;
    const unsigned dd  = tmp % 96u;
    const unsigned bb  = tmp / 96u;

    const unsigned h_in = (2u * ho + 1u) / 3u;   // 192 -> 128
    const unsigned d_in = (2u * dd + 1u) / 3u;   // 96  -> 64

    // Input row base in float4 units: row has 128*16 floats = 512 f4.
    const f4* inRow =
        (const f4*)in + (size_t)((bb * 64u + d_in) * 128u + h_in) * 512u;

    async_f4_to_lds(inRow + t,        &tile[t]);
    async_f4_to_lds(inRow + t + 256u, &tile[t + 256u]);
    wait_async_zero();     // this wave's async copies landed in LDS
    __syncthreads();       // all waves' copies landed

    // Output row base in float4 units: row has 192*16 floats = 768 f4.
    f4* outRow = (f4*)out + (size_t)bid * 768u;

#pragma unroll
    for (int r = 0; r < 3; ++r) {
        const unsigned k     = t + (unsigned)r * 256u;  // f4 index in out row
        const unsigned w_out = k >> 2;                  // output voxel (W)
        const unsigned c4    = k & 3u;                  // which f4 of 16 ch
        const unsigned w_in  = (2u * w_out + 1u) / 3u;  // 192 -> 128
        const f4 v = tile[w_in * 4u + c4];              // ds_load_b128
        __builtin_nontemporal_store(v, &outRow[k]);     // NT b128 store
    }
}

extern "C" void kernel_launch(void* const* d_in, const int* in_sizes, int n_in,
                              void* d_out, int out_size, void* d_ws, size_t ws_size,
                              hipStream_t stream) {
    (void)in_sizes; (void)n_in; (void)out_size; (void)d_ws; (void)ws_size;
    const float* in = (const float*)d_in[0];
    float* out      = (float*)d_out;
    const unsigned blocks = 2u * 96u * 192u;  // one per output (b,d,h) row
    Resize3D_52304111731438_kernel<<<blocks, 256, 0, stream>>>(in, out);
}